// EMRoutingLayer_9775345565737
// MI455X (gfx1250) — compile-verified
//
#include <hip/hip_runtime.h>
#include <stdint.h>

// EM routing, fused single-HBM-pass version for MI455X (gfx1250).
// One workgroup per batch row; 40KB row staged to LDS via async DMA
// (global_load_async_to_lds_b128, ASYNCcnt); the row is read from LDS exactly
// once into registers. Iter-1 reduction, then exact two-phase softmax
// (block max, then one exp per element) fused with the iter-2 reduction.

#define NTHREADS 256
#define NELEM    2048
#define NCH      5
#define EPT      (NELEM / NTHREADS)   // 8 elements per thread
#define NWAVES   (NTHREADS / 32)      // 8 wave32 waves
#define EPSV     1e-8f

#if defined(__gfx1250__) && __has_builtin(__builtin_amdgcn_global_load_async_to_lds_b128)
#define HAVE_ASYNC_LDS 1
#else
#define HAVE_ASYNC_LDS 0
#endif

typedef int v4i __attribute__((ext_vector_type(4)));
typedef __attribute__((address_space(1))) v4i* gptr4;   // global int4*
typedef __attribute__((address_space(3))) v4i* lptr4;   // LDS int4*

__launch_bounds__(NTHREADS)
__global__ void em_routing_kernel(const float* __restrict__ in,
                                  float* __restrict__ out) {
  __shared__ float tile[NELEM * NCH];   // 40 KB row staging
  __shared__ float red[NWAVES][8];      // cross-wave reduction scratch
  __shared__ float bc[4];               // broadcast pose1

  const int tid  = (int)threadIdx.x;
  const int lane = tid & 31;
  const int wv   = tid >> 5;
  const float* row = in + (size_t)blockIdx.x * (size_t)(NELEM * NCH);

  // ---- Stage row (2560 float4 chunks) global -> LDS via async DMA ----
#pragma unroll
  for (int k = 0; k < (NELEM * NCH) / (4 * NTHREADS); ++k) {  // 10 iters
    const int c = (tid + k * NTHREADS) * 4;
#if HAVE_ASYNC_LDS
    gptr4 g = (gptr4)(uintptr_t)(row + c);
    lptr4 l = (lptr4)(uint32_t)(uintptr_t)(&tile[c]);
    __builtin_amdgcn_global_load_async_to_lds_b128(g, l, 0, 0);
#else
    *(float4*)(&tile[c]) = *(const float4*)(row + c);
#endif
  }
#if HAVE_ASYNC_LDS
#if __has_builtin(__builtin_amdgcn_s_wait_asynccnt)
  __builtin_amdgcn_s_wait_asynccnt(0);
#else
  asm volatile("s_wait_asynccnt 0" ::: "memory");
#endif
#endif
  __syncthreads();

  // ---- Pass A: read LDS once into registers; iteration 1 with r = 1/N ----
  // pose1 = sum(a*p) / (sum(a) + N*eps)   (the 1/N cancels)
  float P0[EPT], P1[EPT], P2[EPT], P3[EPT], AV[EPT];
  float sa = 0.f;
  float sp[4] = {0.f, 0.f, 0.f, 0.f};
#pragma unroll
  for (int i = 0; i < EPT; ++i) {
    const int e = tid + i * NTHREADS;        // lane stride 5 dwords: bank-conflict free
    const float* p = &tile[e * NCH];
    P0[i] = p[0]; P1[i] = p[1]; P2[i] = p[2]; P3[i] = p[3]; AV[i] = p[4];
    const float a = AV[i];
    sa += a;
    sp[0] += a * P0[i]; sp[1] += a * P1[i]; sp[2] += a * P2[i]; sp[3] += a * P3[i];
  }
#pragma unroll
  for (int s = 16; s >= 1; s >>= 1) {
    sa    += __shfl_xor(sa,    s, 32);
    sp[0] += __shfl_xor(sp[0], s, 32);
    sp[1] += __shfl_xor(sp[1], s, 32);
    sp[2] += __shfl_xor(sp[2], s, 32);
    sp[3] += __shfl_xor(sp[3], s, 32);
  }
  if (lane == 0) {
    red[wv][0] = sa;
    red[wv][1] = sp[0]; red[wv][2] = sp[1]; red[wv][3] = sp[2]; red[wv][4] = sp[3];
  }
  __syncthreads();
  if (tid == 0) {
    float ta = 0.f, t0 = 0.f, t1 = 0.f, t2 = 0.f, t3 = 0.f;
    for (int w = 0; w < NWAVES; ++w) {
      ta += red[w][0]; t0 += red[w][1]; t1 += red[w][2]; t2 += red[w][3]; t3 += red[w][4];
    }
    const float inv = 1.f / (ta + EPSV * (float)NELEM);
    bc[0] = t0 * inv; bc[1] = t1 * inv; bc[2] = t2 * inv; bc[3] = t3 * inv;
  }
  __syncthreads();
  const float c0 = bc[0], c1 = bc[1], c2 = bc[2], c3 = bc[3];

  // ---- Phase 1 of softmax: nd = -||p - pose1||^2 and the block max ----
  float nd[EPT];
  float mx = -3.0e38f;
#pragma unroll
  for (int i = 0; i < EPT; ++i) {
    const float d0 = P0[i] - c0, d1 = P1[i] - c1, d2 = P2[i] - c2, d3 = P3[i] - c3;
    nd[i] = -(d0 * d0 + d1 * d1 + d2 * d2 + d3 * d3);
    mx = fmaxf(mx, nd[i]);
  }
#pragma unroll
  for (int s = 16; s >= 1; s >>= 1)
    mx = fmaxf(mx, __shfl_xor(mx, s, 32));
  if (lane == 0) red[wv][0] = mx;
  __syncthreads();
  float M = red[0][0];
#pragma unroll
  for (int w = 1; w < NWAVES; ++w) M = fmaxf(M, red[w][0]);  // LDS broadcast reads
  __syncthreads();  // all reads of red done before it is rewritten below

  // ---- Phase 2 + iteration-2 reduction: one exp per element,
  //      Z = sum e^(nd-M), A = sum e*act, Q = sum e*act*pose ----
  float zz = 0.f, aw = 0.f;
  float q[4] = {0.f, 0.f, 0.f, 0.f};
#pragma unroll
  for (int i = 0; i < EPT; ++i) {
    const float t  = __expf(nd[i] - M);
    const float ta = t * AV[i];
    zz += t;
    aw += ta;
    q[0] += ta * P0[i];
    q[1] += ta * P1[i];
    q[2] += ta * P2[i];
    q[3] += ta * P3[i];
  }
#pragma unroll
  for (int s = 16; s >= 1; s >>= 1) {
    zz   += __shfl_xor(zz,   s, 32);
    aw   += __shfl_xor(aw,   s, 32);
    q[0] += __shfl_xor(q[0], s, 32);
    q[1] += __shfl_xor(q[1], s, 32);
    q[2] += __shfl_xor(q[2], s, 32);
    q[3] += __shfl_xor(q[3], s, 32);
  }
  if (lane == 0) {
    red[wv][0] = zz; red[wv][1] = aw;
    red[wv][2] = q[0]; red[wv][3] = q[1]; red[wv][4] = q[2]; red[wv][5] = q[3];
  }
  __syncthreads();
  if (tid == 0) {
    float Z = 0.f, A = 0.f, Q0 = 0.f, Q1 = 0.f, Q2 = 0.f, Q3 = 0.f;
    for (int w = 0; w < NWAVES; ++w) {
      Z  += red[w][0];
      A  += red[w][1];
      Q0 += red[w][2];
      Q1 += red[w][3];
      Q2 += red[w][4];
      Q3 += red[w][5];
    }
    // pose2 = (Q/Z) / (A/Z + eps) = Q / (A + eps*Z);  sum_weights = A/Z
    const float inv = 1.f / (A + EPSV * Z);
    float* o = out + (size_t)blockIdx.x * NCH;
    o[0] = Q0 * inv;
    o[1] = Q1 * inv;
    o[2] = Q2 * inv;
    o[3] = Q3 * inv;
    o[4] = A / Z;
  }
}

extern "C" void kernel_launch(void* const* d_in, const int* in_sizes, int n_in,
                              void* d_out, int out_size, void* d_ws, size_t ws_size,
                              hipStream_t stream) {
  (void)n_in; (void)d_ws; (void)ws_size; (void)out_size;
  const float* in = (const float*)d_in[0];
  float* out = (float*)d_out;
  const int B = in_sizes[0] / (NELEM * NCH);   // 4096
  em_routing_kernel<<<B, NTHREADS, 0, stream>>>(in, out);
}